// GCN_SSF_29669634081190
// MI455X (gfx1250) — compile-verified
//
#include <hip/hip_runtime.h>
#include <math.h>

typedef __attribute__((ext_vector_type(2))) float v2f;
typedef __attribute__((ext_vector_type(4))) float v4f;
typedef __attribute__((ext_vector_type(8))) float v8f;

#define NFEAT  512
#define NHID   512
#define NSSF   256
#define NCLASS 64

// ---------------------------------------------------------------------------
// Device-scope f32 atomic add (GLOBAL_ATOMIC_ADD_F32, no return -> STOREcnt).
// ---------------------------------------------------------------------------
__device__ __forceinline__ void atomic_add_f32_dev(float* p, float v) {
  asm volatile("global_atomic_add_f32 %0, %1, off scope:SCOPE_DEV"
               :: "v"(p), "v"(v) : "memory");
}

// ---------------------------------------------------------------------------
// Zero fill (float4 granularity).
// ---------------------------------------------------------------------------
__global__ void fill_zero4(float* __restrict__ p, long n4) {
  long i = (long)blockIdx.x * blockDim.x + threadIdx.x;
  if (i < n4) {
    v4f z = {};
    ((v4f*)p)[i] = z;
  }
}

// ---------------------------------------------------------------------------
// C[M,N] = act(A)[M,K] @ W[K,N] + bias   via V_WMMA_F32_16X16X4_F32.
// One wave per 16x64 C strip (4 WMMA accumulators share one A-fragment):
// per K-step of 4 we issue 1x global_load_b128 (A), 8x global_load_b32 with
// immediate offsets off two bumped B-row pointers, and 4 WMMAs.
// Requires M%16==0, N%64==0, K%4==0.
// A-frag (16x4 f32): lanes 0-15 hold (K0,K1) of row M=lane; lanes 16-31 hold
// (K2,K3) of row M=lane-16.  B-frag (4x16): VGPR0 = row K0 (lanes 0-15) /
// row K2 (lanes 16-31); VGPR1 = K1 / K3.  C/D: VGPR r = row r (lanes 0-15)
// and row r+8 (lanes 16-31), column = lane&15.
// ---------------------------------------------------------------------------
template <bool RELU_IN>
__global__ void __launch_bounds__(256)
gemm_bias_wmma(const float* __restrict__ A, const float* __restrict__ W,
               const float* __restrict__ bias, float* __restrict__ C,
               int M, int N, int K) {
  const int lane    = threadIdx.x & 31;
  const int wave    = threadIdx.x >> 5;
  const int stripsN = N >> 6;                      // 16x64 strips along N
  const long stripsM = (long)(M >> 4);
  const long strip = (long)blockIdx.x * (blockDim.x >> 5) + wave;
  if (strip >= stripsM * (long)stripsN) return;    // whole-wave uniform exit

  const long sM  = strip / stripsN;
  const int  sN  = (int)(strip - sM * stripsN);
  const int  r16 = lane & 15;
  const int  hi  = lane >> 4;                      // 0: K0/K1 half, 1: K2/K3

  const long rowA = sM * 16 + r16;
  const int  colb = (sN << 6) + r16;               // strip base column + lane

  const float* __restrict__ pA  = A + rowA * (long)K;
  const float* __restrict__ pB0 = W + (long)(hi << 1) * N + colb; // row k+2*hi
  const float* __restrict__ pB1 = pB0 + N;                        // row +1
  const long strideB = 4L * N;

  v8f acc0 = {}, acc1 = {}, acc2 = {}, acc3 = {};

  for (int k = 0; k < K; k += 4) {
    if ((k & 63) == 0)
      __builtin_prefetch(pA + 256, 0, 3);

    v4f a4 = *(const v4f*)pA;
    pA += 4;
    if (RELU_IN) {
      a4.x = fmaxf(a4.x, 0.0f); a4.y = fmaxf(a4.y, 0.0f);
      a4.z = fmaxf(a4.z, 0.0f); a4.w = fmaxf(a4.w, 0.0f);
    }
    v2f a;
    a.x = hi ? a4.z : a4.x;
    a.y = hi ? a4.w : a4.y;

    v2f b0, b1, b2, b3;
    b0.x = pB0[0];  b0.y = pB1[0];
    b1.x = pB0[16]; b1.y = pB1[16];
    b2.x = pB0[32]; b2.y = pB1[32];
    b3.x = pB0[48]; b3.y = pB1[48];
    pB0 += strideB;
    pB1 += strideB;

    acc0 = __builtin_amdgcn_wmma_f32_16x16x4_f32(false, a, false, b0,
                                                 (short)0, acc0, false, false);
    acc1 = __builtin_amdgcn_wmma_f32_16x16x4_f32(false, a, false, b1,
                                                 (short)0, acc1, false, false);
    acc2 = __builtin_amdgcn_wmma_f32_16x16x4_f32(false, a, false, b2,
                                                 (short)0, acc2, false, false);
    acc3 = __builtin_amdgcn_wmma_f32_16x16x4_f32(false, a, false, b3,
                                                 (short)0, acc3, false, false);
  }

  const float bv0 = bias ? bias[colb]      : 0.0f;
  const float bv1 = bias ? bias[colb + 16] : 0.0f;
  const float bv2 = bias ? bias[colb + 32] : 0.0f;
  const float bv3 = bias ? bias[colb + 48] : 0.0f;

  float* __restrict__ Cp = C + (sM * 16 + (long)hi * 8) * (long)N + colb;
#pragma unroll
  for (int r = 0; r < 8; ++r) {
    float* row = Cp + (long)r * N;
    row[0]  = acc0[r] + bv0;
    row[16] = acc1[r] + bv1;
    row[32] = acc2[r] + bv2;
    row[48] = acc3[r] + bv3;
  }
}

// ---------------------------------------------------------------------------
// SPMM scatter: out[dst[e], :] += m[src[e], :] * vals[e]
// One thread per (edge, 4-feature chunk). F % 4 == 0, 256 % (F/4) == 0.
// ---------------------------------------------------------------------------
__global__ void __launch_bounds__(256)
spmm_scatter(const float* __restrict__ m, const int* __restrict__ src,
             const int* __restrict__ dst, const float* __restrict__ vals,
             float* __restrict__ out, int F, long nEdges) {
  const int chunks = F >> 2;
  const int epb    = 256 / chunks;
  const int le     = threadIdx.x / chunks;
  const int c      = threadIdx.x - le * chunks;
  const long e     = (long)blockIdx.x * epb + le;
  if (e >= nEdges) return;

  const int   s = src[e];
  const int   d = dst[e];
  const float v = vals[e];

  v4f mv = *(const v4f*)(m + (long)s * F + (c << 2));
  float* o = out + (long)d * F + (c << 2);
  atomic_add_f32_dev(o + 0, mv.x * v);
  atomic_add_f32_dev(o + 1, mv.y * v);
  atomic_add_f32_dev(o + 2, mv.z * v);
  atomic_add_f32_dev(o + 3, mv.w * v);
}

// ---------------------------------------------------------------------------
// Exact k-th order statistic of |ssf| via binary search on uint bit pattern.
// Single block. thr = sort(|ssf|)[kth] (ascending, 0-indexed).
// ---------------------------------------------------------------------------
__global__ void __launch_bounds__(256)
select_threshold(const float* __restrict__ ssf, int n, int kth,
                 float* __restrict__ thr) {
  __shared__ unsigned cnt[256];
  __shared__ unsigned s_lo, s_hi;
  if (threadIdx.x == 0) { s_lo = 0u; s_hi = 0x7FFFFFFFu; }
  __syncthreads();

  for (int it = 0; it < 32; ++it) {
    unsigned lo = s_lo, hi = s_hi;
    if (lo >= hi) break;                         // uniform (shared-sourced)
    unsigned mid = lo + ((hi - lo) >> 1);
    unsigned c = 0;
    for (int i = threadIdx.x; i < n; i += 256) {
      unsigned key = __float_as_uint(ssf[i]) & 0x7FFFFFFFu;
      c += (key <= mid) ? 1u : 0u;
    }
    cnt[threadIdx.x] = c;
    __syncthreads();
    for (int s = 128; s > 0; s >>= 1) {
      if ((int)threadIdx.x < s) cnt[threadIdx.x] += cnt[threadIdx.x + s];
      __syncthreads();
    }
    if (threadIdx.x == 0) {
      if (cnt[0] >= (unsigned)(kth + 1)) s_hi = mid; else s_lo = mid + 1;
    }
    __syncthreads();
  }
  if (threadIdx.x == 0) *thr = __uint_as_float(s_lo);
}

// ---------------------------------------------------------------------------
// ssf_sp = where(|ssf| >= thr, ssf, 0); per-class colsq and clamped col norm.
// One block per class column (64 blocks x 256 rows).
// ---------------------------------------------------------------------------
__global__ void __launch_bounds__(256)
ssf_sparsify(const float* __restrict__ ssf, const float* __restrict__ thr,
             float* __restrict__ ssf_sp, float* __restrict__ colsq,
             float* __restrict__ cn) {
  __shared__ float red[256];
  const int c = blockIdx.x;
  const int r = threadIdx.x;
  const float t = *thr;
  const float v = ssf[r * NCLASS + c];
  const float sp = (fabsf(v) >= t) ? v : 0.0f;
  ssf_sp[r * NCLASS + c] = sp;
  red[r] = sp * sp;
  __syncthreads();
  for (int s = 128; s > 0; s >>= 1) {
    if (r < s) red[r] += red[r + s];
    __syncthreads();
  }
  if (r == 0) {
    colsq[c] = red[0];
    cn[c] = fmaxf(sqrtf(red[0]), 1e-6f);
  }
}

// ---------------------------------------------------------------------------
// hsq[row] = sum(h[row,:]^2), h is [M, NSSF]. One wave per row.
// ---------------------------------------------------------------------------
__global__ void __launch_bounds__(256)
row_sumsq(const float* __restrict__ h, float* __restrict__ hsq, int M) {
  const int lane = threadIdx.x & 31;
  const int wave = threadIdx.x >> 5;
  const long row = (long)blockIdx.x * 8 + wave;
  if (row >= M) return;
  const float* hr = h + row * (long)NSSF;
  float s = 0.0f;
#pragma unroll
  for (int j = 0; j < NSSF; j += 32) {
    float x = hr[j + lane];
    s += x * x;
  }
  for (int m = 16; m > 0; m >>= 1) s += __shfl_xor(s, m, 32);
  if (lane == 0) hsq[row] = s;
}

// ---------------------------------------------------------------------------
// Loss: 0.5*(log_softmax(-sqrt(max(hsq-2*out+colsq,1e-12)))
//          + log_softmax(out/(hn*cn)))   per row over 64 classes.
// One wave per row; lane handles classes {lane, lane+32}.
// ---------------------------------------------------------------------------
__global__ void __launch_bounds__(256)
loss_kernel(const float* __restrict__ out, const float* __restrict__ hsq,
            const float* __restrict__ colsq, const float* __restrict__ cn,
            float* __restrict__ loss, int M) {
  const int lane = threadIdx.x & 31;
  const int wave = threadIdx.x >> 5;
  const long row = (long)blockIdx.x * 8 + wave;
  if (row >= M) return;

  const float hq = hsq[row];
  const float hn = fmaxf(sqrtf(hq), 1e-6f);
  const float o0 = out[row * NCLASS + lane];
  const float o1 = out[row * NCLASS + lane + 32];

  const float d0 = -sqrtf(fmaxf(hq - 2.0f * o0 + colsq[lane], 1e-12f));
  const float d1 = -sqrtf(fmaxf(hq - 2.0f * o1 + colsq[lane + 32], 1e-12f));
  const float s0 = o0 / (hn * cn[lane]);
  const float s1 = o1 / (hn * cn[lane + 32]);

  float md = fmaxf(d0, d1);
  float ms = fmaxf(s0, s1);
  for (int m = 16; m > 0; m >>= 1) {
    md = fmaxf(md, __shfl_xor(md, m, 32));
    ms = fmaxf(ms, __shfl_xor(ms, m, 32));
  }
  float ed = expf(d0 - md) + expf(d1 - md);
  float es = expf(s0 - ms) + expf(s1 - ms);
  for (int m = 16; m > 0; m >>= 1) {
    ed += __shfl_xor(ed, m, 32);
    es += __shfl_xor(es, m, 32);
  }
  const float ld = logf(ed);
  const float ls = logf(es);
  loss[row * NCLASS + lane]      = 0.5f * ((d0 - md - ld) + (s0 - ms - ls));
  loss[row * NCLASS + lane + 32] = 0.5f * ((d1 - md - ld) + (s1 - ms - ls));
}

__global__ void copy_scalar(const float* __restrict__ s, float* __restrict__ d) {
  if (threadIdx.x == 0 && blockIdx.x == 0) d[0] = s[0];
}

// ---------------------------------------------------------------------------
// Host launcher.
// ---------------------------------------------------------------------------
extern "C" void kernel_launch(void* const* d_in, const int* in_sizes, int n_in,
                              void* d_out, int out_size, void* d_ws, size_t ws_size,
                              hipStream_t stream) {
  const float* x     = (const float*)d_in[0];
  const int*   src   = (const int*)d_in[1];
  const int*   dst   = (const int*)d_in[2];
  const float* vals  = (const float*)d_in[3];
  const float* W1    = (const float*)d_in[4];
  const float* b1    = (const float*)d_in[5];
  const float* W2    = (const float*)d_in[6];
  const float* b2    = (const float*)d_in[7];
  const float* W3    = (const float*)d_in[8];
  const float* b3    = (const float*)d_in[9];
  const float* ssf   = (const float*)d_in[10];
  const float* sigma = (const float*)d_in[11];

  const long N = (long)in_sizes[0] / NFEAT;   // 50000
  const long E = (long)in_sizes[1];           // 1600000

  // Output layout (reference tuple flattened in order):
  float* out_o  = (float*)d_out;               // [N, 64]
  float* ssf_o  = out_o + N * NCLASS;          // [256, 64]
  float* h_o    = ssf_o + (long)NSSF * NCLASS; // [N, 256]
  float* loss_o = h_o + N * NSSF;              // [N, 64]
  float* sig_o  = loss_o + N * NCLASS;         // [1]

  // Workspace layout.
  float* bufA  = (float*)d_ws;       // [N, 512]
  float* bufB  = bufA + N * NHID;    // [N, 512]
  float* hsq   = bufB + N * NHID;    // [N]
  float* thr   = hsq + N;            // [1]
  float* colsq = thr + 1;            // [64]
  float* cn    = colsq + NCLASS;     // [64]

  auto zero = [&](float* p, long n) {
    long n4 = n >> 2;
    int blocks = (int)((n4 + 255) / 256);
    fill_zero4<<<blocks, 256, 0, stream>>>(p, n4);
  };
  auto gemm = [&](const float* A, const float* W, const float* b, float* C,
                  long M, int Nd, int K, bool relu) {
    long strips = (M >> 4) * (long)(Nd >> 6);   // 16x64 strips
    int blocks = (int)((strips + 7) / 8);
    if (relu)
      gemm_bias_wmma<true><<<blocks, 256, 0, stream>>>(A, W, b, C, (int)M, Nd, K);
    else
      gemm_bias_wmma<false><<<blocks, 256, 0, stream>>>(A, W, b, C, (int)M, Nd, K);
  };
  auto spmm = [&](const float* m, float* o, int F) {
    int epb = 256 / (F >> 2);
    int blocks = (int)((E + epb - 1) / epb);
    spmm_scatter<<<blocks, 256, 0, stream>>>(m, src, dst, vals, o, F, E);
  };

  // --- ssf sparsification chain (independent of node features) ---
  select_threshold<<<1, 256, 0, stream>>>(ssf, NSSF * NCLASS, 8192, thr);
  ssf_sparsify<<<NCLASS, 256, 0, stream>>>(ssf, thr, ssf_o, colsq, cn);

  // --- GCN layer 1: h1 = spmm(x @ W1 + b1) ---
  gemm(x, W1, b1, bufA, N, NHID, NFEAT, false);
  zero(bufB, N * NHID);
  spmm(bufA, bufB, NHID);

  // --- GCN layer 2: h2 = spmm(relu(h1) @ W2 + b2) ---
  gemm(bufB, W2, b2, bufA, N, NHID, NHID, true);
  zero(bufB, N * NHID);
  spmm(bufA, bufB, NHID);

  // --- GCN layer 3: h = spmm(relu(h2) @ W3 + b3) ---
  gemm(bufB, W3, b3, bufA, N, NSSF, NHID, true);
  zero(h_o, N * NSSF);
  spmm(bufA, h_o, NSSF);

  // --- epilogue ---
  row_sumsq<<<(int)((N + 7) / 8), 256, 0, stream>>>(h_o, hsq, (int)N);
  gemm(h_o, ssf_o, nullptr, out_o, N, NCLASS, NSSF, false);
  loss_kernel<<<(int)((N + 7) / 8), 256, 0, stream>>>(out_o, hsq, colsq, cn,
                                                      loss_o, (int)N);
  copy_scalar<<<1, 32, 0, stream>>>(sigma, sig_o);
}